// Model_60765197304061
// MI455X (gfx1250) — compile-verified
//
#include <hip/hip_runtime.h>

typedef float v2f __attribute__((ext_vector_type(2)));
typedef float v8f __attribute__((ext_vector_type(8)));

#define B_SZ    64
#define N_ATOM  256
#define L_PROT  2048
#define DIM     128

// ---------------------------------------------------------------------------
// Embedding gather: out[row, d] = emb[idx[row], d], DIM = 128
// ---------------------------------------------------------------------------
__global__ __launch_bounds__(256) void gather_kernel(const int* __restrict__ idx,
                                                     const float* __restrict__ emb,
                                                     float* __restrict__ out,
                                                     int n_rows) {
  int t = blockIdx.x * 256 + threadIdx.x;
  int row = t >> 7, d = t & 127;
  if (row < n_rows) out[t] = emb[(size_t)idx[row] * DIM + d];
}

// ---------------------------------------------------------------------------
// Y = [relu](X @ W^T + bias).  X: MxK row-major, W: NxK row-major.
// One wave -> one 16x16 tile via V_WMMA_F32_16X16X4_F32.
// Block = 8 waves covering 128 consecutive output columns; A tile in LDS.
// grid = (M/16, N/128). K multiple of 4, K <= 256.
// ---------------------------------------------------------------------------
__global__ __launch_bounds__(256) void dense_wmma_kernel(
    const float* __restrict__ X, const float* __restrict__ W,
    const float* __restrict__ bias, float* __restrict__ Y,
    int K, int N, int do_relu) {
  __shared__ float sA[16 * 256];
  const int m0   = blockIdx.x * 16;
  const int n0   = blockIdx.y * 128 + (threadIdx.x >> 5) * 16;
  const int lane = threadIdx.x & 31;

  // Cooperative stage of the 16xK A tile
  const int tile_elems = 16 * K;
  for (int idx = threadIdx.x; idx < tile_elems; idx += 256) {
    int r = idx / K, c = idx - r * K;
    sA[idx] = X[(size_t)(m0 + r) * K + c];
  }
  __syncthreads();

  v8f acc = {};
  const int kk = (lane >> 4) << 1;   // 0 for lanes 0-15, 2 for lanes 16-31
  const int nr = lane & 15;          // A row index / B column index
  for (int k0 = 0; k0 < K; k0 += 4) {
    v2f a, b;
    a.x = sA[nr * K + k0 + kk];
    a.y = sA[nr * K + k0 + kk + 1];
    const float* wr = W + (size_t)(n0 + nr) * K + k0 + kk;  // B[k][n] = W[n][k]
    b.x = wr[0];
    b.y = wr[1];
    acc = __builtin_amdgcn_wmma_f32_16x16x4_f32(false, a, false, b,
                                                (short)0, acc, false, false);
  }

  const int n  = n0 + nr;
  const int mb = m0 + ((lane >> 4) << 3);
  const float bv = bias ? bias[n] : 0.f;
#pragma unroll
  for (int v = 0; v < 8; ++v) {
    float y = acc[v] + bv;
    if (do_relu) y = fmaxf(y, 0.f);
    Y[(size_t)(mb + v) * N + n] = y;
  }
}

// ---------------------------------------------------------------------------
// Xs[b] += Adj[b] (256x256) @ Hs[b] (256x128).  grid = (B, 256/16), block 256.
// ---------------------------------------------------------------------------
__global__ __launch_bounds__(256) void adj_wmma_kernel(const float* __restrict__ Adj,
                                                       const float* __restrict__ Hs,
                                                       float* __restrict__ Xs) {
  __shared__ float sA[16 * 256];
  const int b  = blockIdx.x;
  const int m0 = blockIdx.y * 16;
  const float* A  = Adj + (size_t)b * N_ATOM * N_ATOM;
  const float* Bm = Hs  + (size_t)b * N_ATOM * DIM;
  float*       C  = Xs  + (size_t)b * N_ATOM * DIM;
  const int lane = threadIdx.x & 31;
  const int n0   = (threadIdx.x >> 5) * 16;

  for (int idx = threadIdx.x; idx < 16 * 256; idx += 256) {
    int r = idx >> 8, c = idx & 255;
    sA[idx] = A[(m0 + r) * N_ATOM + c];
  }
  __syncthreads();

  v8f acc = {};
  const int kk = (lane >> 4) << 1;
  const int nr = lane & 15;
  for (int k0 = 0; k0 < 256; k0 += 4) {
    v2f a, bb;
    a.x = sA[nr * 256 + k0 + kk];
    a.y = sA[nr * 256 + k0 + kk + 1];
    bb.x = Bm[(k0 + kk) * DIM + n0 + nr];      // B row-major (k x n)
    bb.y = Bm[(k0 + kk + 1) * DIM + n0 + nr];
    acc = __builtin_amdgcn_wmma_f32_16x16x4_f32(false, a, false, bb,
                                                (short)0, acc, false, false);
  }

  const int n  = n0 + nr;
  const int mb = m0 + ((lane >> 4) << 3);
#pragma unroll
  for (int v = 0; v < 8; ++v)
    C[(mb + v) * DIM + n] += acc[v];
}

// ---------------------------------------------------------------------------
// 11x11 same-pad conv (1 channel) + bias + relu.  grid = (B, 2048/16).
// 26x138 haloed input tile + 121 weights in LDS; 121 fused FMAs per pixel.
// ---------------------------------------------------------------------------
__global__ __launch_bounds__(256) void conv_kernel(const float* __restrict__ In,
                                                   const float* __restrict__ Wt,
                                                   const float* __restrict__ bias,
                                                   float* __restrict__ Out) {
  __shared__ float sIn[26 * 144];
  __shared__ float sW[121];
  const int b  = blockIdx.x;
  const int y0 = blockIdx.y * 16;
  const float* inb = In + (size_t)b * L_PROT * DIM;

  if (threadIdx.x < 121) sW[threadIdx.x] = Wt[threadIdx.x];
  for (int idx = threadIdx.x; idx < 26 * 138; idx += 256) {
    int ry = idx / 138, rx = idx - ry * 138;
    int gy = y0 + ry - 5, gx = rx - 5;
    float v = 0.f;
    if (gy >= 0 && gy < L_PROT && gx >= 0 && gx < DIM) v = inb[gy * DIM + gx];
    sIn[ry * 144 + rx] = v;
  }
  __syncthreads();

  const float bv = bias[0];
  for (int p = threadIdx.x; p < 16 * 128; p += 256) {
    int ly = p >> 7, lx = p & 127;
    float acc = bv;
#pragma unroll
    for (int ky = 0; ky < 11; ++ky)
#pragma unroll
      for (int kx = 0; kx < 11; ++kx)
        acc = fmaf(sIn[(ly + ky) * 144 + lx + kx], sW[ky * 11 + kx], acc);
    Out[(size_t)b * L_PROT * DIM + (y0 + ly) * DIM + lx] = fmaxf(acc, 0.f);
  }
}

// ---------------------------------------------------------------------------
// out[b,d] = sum_l X[b,l,d]*mask[b,l] / sum_l mask[b,l].  grid=B, block=128.
// ---------------------------------------------------------------------------
__global__ __launch_bounds__(128) void pool_mean_kernel(const float* __restrict__ X,
                                                        const float* __restrict__ mask,
                                                        float* __restrict__ out, int L) {
  int b = blockIdx.x, d = threadIdx.x;
  const float* xb = X + (size_t)b * L * DIM;
  const float* mb = mask + (size_t)b * L;
  float s = 0.f, ms = 0.f;
  for (int l = 0; l < L; ++l) { float m = mb[l]; s = fmaf(xb[l * DIM + d], m, s); ms += m; }
  out[b * DIM + d] = s / ms;
}

// wp[b,l] = tanh( hs[b,l,:] . h[b,:] ) * pmask[b,l]
__global__ __launch_bounds__(256) void attn_w_kernel(const float* __restrict__ Hs,
                                                     const float* __restrict__ H,
                                                     const float* __restrict__ pmask,
                                                     float* __restrict__ wp) {
  int t = blockIdx.x * 256 + threadIdx.x;       // t = b*2048 + l
  int b = t >> 11;
  const float4* hs4 = reinterpret_cast<const float4*>(Hs + (size_t)t * DIM);
  const float4* h4  = reinterpret_cast<const float4*>(H + b * DIM);
  float s = 0.f;
#pragma unroll 4
  for (int d = 0; d < DIM / 4; ++d) {
    float4 a = hs4[d], c = h4[d];
    s = fmaf(a.x, c.x, s); s = fmaf(a.y, c.y, s);
    s = fmaf(a.z, c.z, s); s = fmaf(a.w, c.w, s);
  }
  wp[t] = tanhf(s) * pmask[t];
}

// prot[b,d] = sum_l wp[b,l]*hs[b,l,d] / sum_l pmask[b,l].  grid=B, block=128.
__global__ __launch_bounds__(128) void attn_pool_kernel(const float* __restrict__ Hs,
                                                        const float* __restrict__ wp,
                                                        const float* __restrict__ pmask,
                                                        float* __restrict__ prot) {
  int b = blockIdx.x, d = threadIdx.x;
  const float* hb = Hs + (size_t)b * L_PROT * DIM;
  float s = 0.f, ms = 0.f;
  for (int l = 0; l < L_PROT; ++l) {
    s = fmaf(hb[l * DIM + d], wp[b * L_PROT + l], s);
    ms += pmask[b * L_PROT + l];
  }
  prot[b * DIM + d] = s / ms;
}

__global__ __launch_bounds__(256) void concat_kernel(const float* __restrict__ c,
                                                     const float* __restrict__ p,
                                                     float* __restrict__ cat) {
  int t = blockIdx.x * 256 + threadIdx.x;       // t = b*256 + d
  int b = t >> 8, d = t & 255;
  cat[t] = (d < DIM) ? c[b * DIM + d] : p[b * DIM + (d - DIM)];
}

// affinity[b] = cat2[b,:] . Wi + bi ; out = [loss, affinity...]
__global__ __launch_bounds__(64) void final_kernel(const float* __restrict__ cat2,
                                                   const float* __restrict__ Wi,
                                                   const float* __restrict__ bi,
                                                   const float* __restrict__ label,
                                                   float* __restrict__ out) {
  __shared__ float se[B_SZ];
  int b = threadIdx.x;
  float s = bi[0];
  for (int d = 0; d < 2 * DIM; ++d) s = fmaf(cat2[b * 2 * DIM + d], Wi[d], s);
  out[1 + b] = s;
  float e = s - label[b];
  se[b] = e * e;
  __syncthreads();
  if (b == 0) {
    float t = 0.f;
    for (int i = 0; i < B_SZ; ++i) t += se[i];
    out[0] = t / (float)B_SZ;
  }
}

// ---------------------------------------------------------------------------
extern "C" void kernel_launch(void* const* d_in, const int* in_sizes, int n_in,
                              void* d_out, int out_size, void* d_ws, size_t ws_size,
                              hipStream_t stream) {
  const int*   fingerprints = (const int*)d_in[0];
  const float* adjacency    = (const float*)d_in[1];
  const int*   words        = (const int*)d_in[2];
  const float* cmask        = (const float*)d_in[3];
  const float* pmask        = (const float*)d_in[4];
  const float* label        = (const float*)d_in[5];
  const float* emb_fp       = (const float*)d_in[6];
  const float* emb_word     = (const float*)d_in[7];
  const float* Wg           = (const float*)d_in[8];
  const float* bg           = (const float*)d_in[9];
  const float* Wc           = (const float*)d_in[10];
  const float* bc           = (const float*)d_in[11];
  const float* Wa           = (const float*)d_in[12];
  const float* ba           = (const float*)d_in[13];
  const float* Wo           = (const float*)d_in[14];
  const float* bo           = (const float*)d_in[15];
  const float* Wi           = (const float*)d_in[16];
  const float* bi           = (const float*)d_in[17];

  float* ws   = (float*)d_ws;
  float* xs   = ws;                                   // B*256*128
  float* hsb  = xs   + (size_t)B_SZ * N_ATOM * DIM;   // B*256*128
  float* p0   = hsb  + (size_t)B_SZ * N_ATOM * DIM;   // B*2048*128
  float* p1   = p0   + (size_t)B_SZ * L_PROT * DIM;   // B*2048*128
  float* cmp  = p1   + (size_t)B_SZ * L_PROT * DIM;   // B*128
  float* hv   = cmp  + B_SZ * DIM;                    // B*128
  float* wp   = hv   + B_SZ * DIM;                    // B*2048
  float* prot = wp   + B_SZ * L_PROT;                 // B*128
  float* cat  = prot + B_SZ * DIM;                    // B*256
  float* cat2 = cat  + B_SZ * 2 * DIM;                // B*256

  // --- GNN ---
  gather_kernel<<<(B_SZ * N_ATOM * DIM) / 256, 256, 0, stream>>>(
      fingerprints, emb_fp, xs, B_SZ * N_ATOM);
  for (int i = 0; i < 3; ++i) {
    dense_wmma_kernel<<<dim3(B_SZ * N_ATOM / 16, 1), 256, 0, stream>>>(
        xs, Wg + (size_t)i * DIM * DIM, bg + i * DIM, hsb, DIM, DIM, 1);
    adj_wmma_kernel<<<dim3(B_SZ, N_ATOM / 16), 256, 0, stream>>>(adjacency, hsb, xs);
  }
  pool_mean_kernel<<<B_SZ, DIM, 0, stream>>>(xs, cmask, cmp, N_ATOM);

  // --- CNN ---
  gather_kernel<<<(B_SZ * L_PROT * DIM) / 256, 256, 0, stream>>>(
      words, emb_word, p0, B_SZ * L_PROT);
  conv_kernel<<<dim3(B_SZ, L_PROT / 16), 256, 0, stream>>>(p0, Wc + 0 * 121, bc + 0, p1);
  conv_kernel<<<dim3(B_SZ, L_PROT / 16), 256, 0, stream>>>(p1, Wc + 1 * 121, bc + 1, p0);
  conv_kernel<<<dim3(B_SZ, L_PROT / 16), 256, 0, stream>>>(p0, Wc + 2 * 121, bc + 2, p1);

  // --- Attention ---
  dense_wmma_kernel<<<dim3(B_SZ / 16, 1), 256, 0, stream>>>(cmp, Wa, ba, hv, DIM, DIM, 1);
  dense_wmma_kernel<<<dim3(B_SZ * L_PROT / 16, 1), 256, 0, stream>>>(
      p1, Wa, ba, p0, DIM, DIM, 1);
  attn_w_kernel<<<B_SZ * L_PROT / 256, 256, 0, stream>>>(p0, hv, pmask, wp);
  attn_pool_kernel<<<B_SZ, DIM, 0, stream>>>(p0, wp, pmask, prot);

  // --- Output MLP ---
  concat_kernel<<<B_SZ * 2 * DIM / 256, 256, 0, stream>>>(cmp, prot, cat);
  dense_wmma_kernel<<<dim3(B_SZ / 16, 2), 256, 0, stream>>>(
      cat, Wo, bo, cat2, 2 * DIM, 2 * DIM, 1);
  dense_wmma_kernel<<<dim3(B_SZ / 16, 2), 256, 0, stream>>>(
      cat2, Wo + 2 * DIM * 2 * DIM, bo + 2 * DIM, cat, 2 * DIM, 2 * DIM, 1);
  final_kernel<<<1, B_SZ, 0, stream>>>(cat, Wi, bi, label, (float*)d_out);
}